// ScaledDotProductAttention_41145786696213
// MI455X (gfx1250) — compile-verified
//
#include <hip/hip_runtime.h>
#include <math.h>
#include <stdint.h>

typedef __attribute__((ext_vector_type(2))) float v2f;
typedef __attribute__((ext_vector_type(8))) float v8f;

#define SEQ 2048
#define DIMV 64
#define NJT (SEQ / 16)
#define KVSTRIDE 68   // padded f32 stride for K/V tiles in LDS (conflict-free column reads)
#define ASTRIDE 17    // padded f32 stride for attn tile in LDS

// One wave (32 lanes) per (batch, 16-row query tile).
// WMMA F32_16X16X4_F32 layouts (wave32):
//   A 16x4 : lane L -> row M = L&15 ; VGPR0 = K {0|2}, VGPR1 = K {1|3} (half = L>>4 adds +2)
//   B 4x16 : lane L -> col N = L&15 ; VGPR0 = K {0|2}, VGPR1 = K {1|3}
//   C/D 16x16 : lane L -> col N = L&15 ; VGPR r -> row M = r + 8*(L>>4)
__launch_bounds__(32, 1)
__global__ void sdpa_wmma_f32_kernel(const float* __restrict__ Q,
                                     const float* __restrict__ K,
                                     const float* __restrict__ V,
                                     const unsigned char* __restrict__ mask,
                                     float* __restrict__ ctx_out,
                                     float* __restrict__ attn_out)
{
    __shared__ float ldsK[2][16 * KVSTRIDE];   // double-buffered K tile
    __shared__ float ldsV[2][16 * KVSTRIDE];   // double-buffered V tile
    __shared__ float ldsA[16 * ASTRIDE];       // attn tile re-layout staging

    const int lane = threadIdx.x;        // 0..31
    const int half = lane >> 4;          // 0 | 1
    const int n    = lane & 15;          // column / A-row index
    const int b    = blockIdx.x / NJT;
    const int it   = blockIdx.x % NJT;
    const int q0   = it * 16;

    // LDS byte addresses for the async-DMA destination operand
    const unsigned ldsKb[2] = { (unsigned)(uintptr_t)&ldsK[0][0],
                                (unsigned)(uintptr_t)&ldsK[1][0] };
    const unsigned ldsVb[2] = { (unsigned)(uintptr_t)&ldsV[0][0],
                                (unsigned)(uintptr_t)&ldsV[1][0] };

    // ---- Q A-fragments with 1/sqrt(D) pre-folded (16 frags of K=4) ----
    v2f aq[16];
    {
        const float* qb = Q + ((size_t)b * SEQ + q0 + n) * DIMV; // row M = n
        #pragma unroll
        for (int kb = 0; kb < 16; ++kb) {
            const int k0 = kb * 4 + half * 2;
            aq[kb].x = qb[k0]     * 0.125f;
            aq[kb].y = qb[k0 + 1] * 0.125f;
        }
    }

    // ---- per-lane row state for the 8 rows it carries in the C/D layout ----
    bool  rowm[8];
    float mrow[8], lrow[8];
    #pragma unroll
    for (int r = 0; r < 8; ++r) {
        rowm[r] = mask[(size_t)b * SEQ + q0 + r + 8 * half] != 0;
        mrow[r] = -INFINITY;
        lrow[r] = 0.0f;
    }

    // Async DMA of a 16x64 f32 tile: 8 x b128 per wave, global -> padded LDS rows.
    // Tracked on ASYNCcnt; no VGPR staging, overlaps with WMMA compute.
    auto async_load_tile = [&](const float* __restrict__ src, unsigned lds_base) {
        #pragma unroll
        for (int i = 0; i < 8; ++i) {
            const int idx = lane + 32 * i;   // 0..255 float4 slots
            const int row = idx >> 4;
            const int c4  = idx & 15;
            const unsigned laddr = lds_base + (unsigned)((row * KVSTRIDE + c4 * 4) * 4);
            const unsigned long long gaddr =
                (unsigned long long)(uintptr_t)(src + (size_t)row * DIMV + c4 * 4);
            asm volatile("global_load_async_to_lds_b128 %0, %1, off"
                         :: "v"(laddr), "v"(gaddr) : "memory");
        }
    };

    // S_tile = (Q/sqrtD) (16x64) @ K_tile^T (64x16) via 16 chained f32 WMMAs
    auto compute_scores = [&](const float* __restrict__ kbuf) -> v8f {
        v8f cs = {0.f, 0.f, 0.f, 0.f, 0.f, 0.f, 0.f, 0.f};
        #pragma unroll
        for (int kb = 0; kb < 16; ++kb) {
            const int k0 = kb * 4 + half * 2;
            v2f bb;
            bb.x = kbuf[n * KVSTRIDE + k0];     // B[k][n] = K[key n][feat k]
            bb.y = kbuf[n * KVSTRIDE + k0 + 1];
            cs = __builtin_amdgcn_wmma_f32_16x16x4_f32(
                     false, aq[kb], false, bb, (short)0, cs, false, false);
        }
        return cs;
    };

    // ========== Pass 1: per-lane online max/exp-sum (no per-tile shuffles) ==========
    async_load_tile(K + (size_t)b * SEQ * DIMV, ldsKb[0]);
    for (int jt = 0; jt < NJT; ++jt) {
        const int cur = jt & 1;
        const int j0  = jt * 16;
        if (jt + 1 < NJT) {
            async_load_tile(K + ((size_t)b * SEQ + (size_t)(jt + 1) * 16) * DIMV,
                            ldsKb[cur ^ 1]);
            asm volatile("s_wait_asynccnt 0x8" ::: "memory");  // current tile landed
        } else {
            asm volatile("s_wait_asynccnt 0x0" ::: "memory");
        }

        const bool colm = mask[(size_t)b * SEQ + j0 + n] != 0;
        v8f cs = compute_scores(&ldsK[cur][0]);

        #pragma unroll
        for (int r = 0; r < 8; ++r) {
            const float s = (colm || rowm[r]) ? -INFINITY : cs[r];
            // single-transcendental online update; inf corner cases fall out:
            //   m=-inf, s finite: d=+inf, e=0, l -> 1 ; s=-inf: d=-inf, e=0, l += 0
            const float d = (s == -INFINITY) ? -INFINITY : s - mrow[r];
            mrow[r] = fmaxf(mrow[r], s);
            const float e = __expf(-fabsf(d));
            lrow[r] = (d > 0.0f) ? (lrow[r] * e + 1.0f) : (lrow[r] + e);
        }
    }

    // one-time 16-lane merge of per-lane (m, l) partials per row
    float invl[8];
    #pragma unroll
    for (int r = 0; r < 8; ++r) {
        float m = mrow[r], l = lrow[r];
        #pragma unroll
        for (int off = 1; off < 16; off <<= 1) {
            const float mo = __shfl_xor(m, off, 32);
            const float lo = __shfl_xor(l, off, 32);
            const float mn = fmaxf(m, mo);
            const float a1 = (m  == -INFINITY) ? 0.0f : __expf(m - mn);
            const float a2 = (mo == -INFINITY) ? 0.0f : __expf(mo - mn);
            l = l * a1 + lo * a2;
            m = mn;
        }
        mrow[r] = m;
        lrow[r] = l;
        invl[r] = (l > 0.0f) ? (1.0f / l) : 0.0f;
    }

    // ====== Pass 2: recompute scores, emit attn (NT), accumulate attn@V ======
    v8f acc[4] = { {0.f,0.f,0.f,0.f,0.f,0.f,0.f,0.f}, {0.f,0.f,0.f,0.f,0.f,0.f,0.f,0.f},
                   {0.f,0.f,0.f,0.f,0.f,0.f,0.f,0.f}, {0.f,0.f,0.f,0.f,0.f,0.f,0.f,0.f} };

    async_load_tile(K + (size_t)b * SEQ * DIMV, ldsKb[0]);
    async_load_tile(V + (size_t)b * SEQ * DIMV, ldsVb[0]);
    for (int jt = 0; jt < NJT; ++jt) {
        const int cur = jt & 1;
        const int j0  = jt * 16;
        if (jt + 1 < NJT) {
            const size_t nb = ((size_t)b * SEQ + (size_t)(jt + 1) * 16) * DIMV;
            async_load_tile(K + nb, ldsKb[cur ^ 1]);
            async_load_tile(V + nb, ldsVb[cur ^ 1]);
            asm volatile("s_wait_asynccnt 0x10" ::: "memory"); // current K+V landed
        } else {
            asm volatile("s_wait_asynccnt 0x0" ::: "memory");
        }

        const bool colm = mask[(size_t)b * SEQ + j0 + n] != 0;
        v8f cs = compute_scores(&ldsK[cur][0]);

        #pragma unroll
        for (int r = 0; r < 8; ++r) {
            const float a = (colm || rowm[r])
                              ? 0.0f
                              : __expf(cs[r] - mrow[r]) * invl[r];
            // 268 MB streaming output, never re-read: non-temporal store
            __builtin_nontemporal_store(
                a, attn_out + ((size_t)b * SEQ + q0 + r + 8 * half) * SEQ + j0 + n);
            ldsA[(r + 8 * half) * ASTRIDE + n] = a;  // stage for A-operand re-layout
        }
        __syncthreads();

        // attn tile as A-fragments (row M = n, K over the 16 keys of this tile)
        v2f aa[4];
        #pragma unroll
        for (int kb = 0; kb < 4; ++kb) {
            const int k0 = kb * 4 + half * 2;
            aa[kb].x = ldsA[n * ASTRIDE + k0];
            aa[kb].y = ldsA[n * ASTRIDE + k0 + 1];
        }

        // context(16x64) += attn(16x16) @ V(16x64): 4 N-tiles x 4 K-steps
        const float* vbuf = &ldsV[cur][0];
        #pragma unroll
        for (int t = 0; t < 4; ++t) {
            #pragma unroll
            for (int kb = 0; kb < 4; ++kb) {
                const int k0 = kb * 4 + half * 2;
                v2f bb;
                bb.x = vbuf[k0 * KVSTRIDE + t * 16 + n];       // B[k][col] = V[key k][col]
                bb.y = vbuf[(k0 + 1) * KVSTRIDE + t * 16 + n];
                acc[t] = __builtin_amdgcn_wmma_f32_16x16x4_f32(
                             false, aa[kb], false, bb, (short)0, acc[t], false, false);
            }
        }
        __syncthreads();   // ldsA reuse guard for next iteration
    }

    // ---- write context [B,S,64] ----
    #pragma unroll
    for (int t = 0; t < 4; ++t)
        #pragma unroll
        for (int r = 0; r < 8; ++r)
            ctx_out[((size_t)b * SEQ + q0 + r + 8 * half) * DIMV + t * 16 + n] = acc[t][r];
}

extern "C" void kernel_launch(void* const* d_in, const int* in_sizes, int n_in,
                              void* d_out, int out_size, void* d_ws, size_t ws_size,
                              hipStream_t stream) {
    const float* Q = (const float*)d_in[0];
    const float* K = (const float*)d_in[1];
    const float* V = (const float*)d_in[2];
    const unsigned char* mask = (const unsigned char*)d_in[3]; // jnp bool_ = 1 byte
    const int B = in_sizes[0] / (SEQ * DIMV);

    float* ctx  = (float*)d_out;                               // [B,S,64]
    float* attn = (float*)d_out + (size_t)B * SEQ * DIMV;      // [B,S,S]

    dim3 grid(B * NJT);   // one wave32 per (batch, 16-query-row tile)
    sdpa_wmma_f32_kernel<<<grid, 32, 0, stream>>>(Q, K, V, mask, ctx, attn);
}